// RoiBoxFeatureExtractor_47193100648834
// MI455X (gfx1250) — compile-verified
//
#include <hip/hip_runtime.h>
#include <hip/hip_bf16.h>
#include <math.h>

// ---------------------------------------------------------------------------
// Problem constants (from reference)
// ---------------------------------------------------------------------------
#define B_     64
#define C_     256
#define H_     100
#define W_     152
#define RES_   7
#define SR_    2
#define SCALE_ 0.0625f
#define EPS_   1e-5f
#define KDIM_  (C_ * RES_ * RES_)   // 12544
#define NDIM_  1024

typedef __attribute__((ext_vector_type(2))) float v2f;
typedef __attribute__((ext_vector_type(8))) float v8f;

// ---------------------------------------------------------------------------
// Kernel 1: RoIAlign (torchvision-style, aligned=false, SR=2)
// One thread per output element [b, c, py, px]. Output layout feat[b][c*49+py*7+px].
// ---------------------------------------------------------------------------
__global__ __launch_bounds__(256) void roi_align_kernel(
    const float* __restrict__ x,      // [B, C, H, W]
    const float* __restrict__ bbox,   // [B, 4] as [xa, xb, ya, yb]
    float* __restrict__ feat)         // [B, KDIM_]
{
    const int idx = blockIdx.x * blockDim.x + threadIdx.x;
    const int total = B_ * C_ * RES_ * RES_;
    if (idx >= total) return;

    const int pp = idx % (RES_ * RES_);
    const int c  = (idx / (RES_ * RES_)) % C_;
    const int b  = idx / (C_ * RES_ * RES_);
    const int py = pp / RES_;
    const int px = pp % RES_;

    // boxes_xyxy = bbox[:, [0,2,1,3]] -> x1 = bbox[0], y1 = bbox[2], x2 = bbox[1], y2 = bbox[3]
    const float x1 = bbox[b * 4 + 0] * SCALE_;
    const float x2 = bbox[b * 4 + 1] * SCALE_;
    const float y1 = bbox[b * 4 + 2] * SCALE_;
    const float y2 = bbox[b * 4 + 3] * SCALE_;

    const float roi_w = fmaxf(x2 - x1, 1.0f);
    const float roi_h = fmaxf(y2 - y1, 1.0f);
    const float bin_w = roi_w / (float)RES_;
    const float bin_h = roi_h / (float)RES_;

    const float* f = x + ((size_t)(b * C_ + c)) * (H_ * W_);

    float acc = 0.0f;
    #pragma unroll
    for (int sy = 0; sy < SR_; ++sy) {
        const float gy = (float)py + ((float)sy + 0.5f) / (float)SR_;
        const float yf = y1 + gy * bin_h;
        #pragma unroll
        for (int sx = 0; sx < SR_; ++sx) {
            const float gx = (float)px + ((float)sx + 0.5f) / (float)SR_;
            const float xf = x1 + gx * bin_w;

            const bool valid = (yf >= -1.0f) && (yf <= (float)H_) &&
                               (xf >= -1.0f) && (xf <= (float)W_);

            const float yc = fminf(fmaxf(yf, 0.0f), (float)(H_ - 1));
            const float xc = fminf(fmaxf(xf, 0.0f), (float)(W_ - 1));
            int y0 = (int)floorf(yc);
            int x0 = (int)floorf(xc);
            y0 = y0 < 0 ? 0 : (y0 > H_ - 1 ? H_ - 1 : y0);
            x0 = x0 < 0 ? 0 : (x0 > W_ - 1 ? W_ - 1 : x0);
            const int y1i = y0 + 1 < H_ ? y0 + 1 : H_ - 1;
            const int x1i = x0 + 1 < W_ ? x0 + 1 : W_ - 1;

            const float ly = yc - (float)y0;
            const float hy = 1.0f - ly;
            const float lx = xc - (float)x0;
            const float hx = 1.0f - lx;

            const float v00 = f[y0  * W_ + x0 ];
            const float v01 = f[y0  * W_ + x1i];
            const float v10 = f[y1i * W_ + x0 ];
            const float v11 = f[y1i * W_ + x1i];

            const float val = hy * (hx * v00 + lx * v01) + ly * (hx * v10 + lx * v11);
            acc += valid ? val : 0.0f;
        }
    }
    feat[(size_t)b * KDIM_ + c * (RES_ * RES_) + pp] = acc * (1.0f / (SR_ * SR_));
}

// ---------------------------------------------------------------------------
// Kernel 2: h = relu(feat @ fc_w.T + fc_b) via V_WMMA_F32_16X16X4_F32.
// One wave32 per 16x16 output tile. grid = (N/16, M/16).
//
// 32-bit A (16x4): lane l holds row M=l&15; lanes 0-15 cover K={k,k+1},
//                  lanes 16-31 cover K={k+2,k+3} (VGPR0=first K, VGPR1=second).
// 32-bit B (4x16): lane l holds col N=l&15 with the same K split.
// B[k][n] = fc_w[n][k] -> both operands read contiguously along K.
// C/D (16x16 f32): VGPR r, lane l -> [M = r + 8*(l>>4)][N = l&15].
// ---------------------------------------------------------------------------
__global__ __launch_bounds__(32) void gemm_wmma_kernel(
    const float* __restrict__ feat,   // [64, KDIM_]
    const float* __restrict__ fc_w,   // [1024, KDIM_]
    const float* __restrict__ fc_b,   // [1024]
    float* __restrict__ h)            // [64, 1024]
{
    const int lane = threadIdx.x & 31;
    const int l16  = lane & 15;
    const int half = lane >> 4;          // 0: K={0,1}; 1: K={2,3}
    const int n0 = blockIdx.x * 16;
    const int m0 = blockIdx.y * 16;

    const float* __restrict__ arow = feat + (size_t)(m0 + l16) * KDIM_ + half * 2;
    const float* __restrict__ brow = fc_w + (size_t)(n0 + l16) * KDIM_ + half * 2;

    v8f acc = {};
    for (int k = 0; k < KDIM_; k += 16) {
        #pragma unroll
        for (int kk = 0; kk < 16; kk += 4) {
            v2f a = *(const v2f*)(arow + k + kk);
            v2f b = *(const v2f*)(brow + k + kk);
            // 8 args: (neg_a, A, neg_b, B, c_mod, C, reuse_a, reuse_b)
            acc = __builtin_amdgcn_wmma_f32_16x16x4_f32(
                false, a, false, b, (short)0, acc, false, false);
        }
    }

    const float bias = fc_b[n0 + l16];
    #pragma unroll
    for (int r = 0; r < 8; ++r) {
        const int m = m0 + r + half * 8;
        float v = acc[r] + bias;
        v = v > 0.0f ? v : 0.0f;
        h[(size_t)m * NDIM_ + n0 + l16] = v;
    }
}

// ---------------------------------------------------------------------------
// Kernel 3: batch-norm across batch axis (64 rows), biased variance like jnp.var.
// One thread per output column.
// ---------------------------------------------------------------------------
__global__ __launch_bounds__(256) void bn_kernel(
    const float* __restrict__ h,      // [64, 1024]
    const float* __restrict__ gamma,
    const float* __restrict__ beta,
    float* __restrict__ out)          // [64, 1024]
{
    const int n = blockIdx.x * blockDim.x + threadIdx.x;
    if (n >= NDIM_) return;

    float s = 0.0f, s2 = 0.0f;
    #pragma unroll 4
    for (int m = 0; m < B_; ++m) {
        const float v = h[(size_t)m * NDIM_ + n];
        s  += v;
        s2 += v * v;
    }
    const float mu  = s * (1.0f / B_);
    const float var = s2 * (1.0f / B_) - mu * mu;
    const float inv = rsqrtf(var + EPS_);
    const float g = gamma[n];
    const float bb = beta[n];

    #pragma unroll 4
    for (int m = 0; m < B_; ++m) {
        out[(size_t)m * NDIM_ + n] = (h[(size_t)m * NDIM_ + n] - mu) * inv * g + bb;
    }
}

// ---------------------------------------------------------------------------
// Launch
// ---------------------------------------------------------------------------
extern "C" void kernel_launch(void* const* d_in, const int* in_sizes, int n_in,
                              void* d_out, int out_size, void* d_ws, size_t ws_size,
                              hipStream_t stream) {
    const float* x     = (const float*)d_in[0];   // [64, 256, 100, 152]
    const float* bbox  = (const float*)d_in[1];   // [64, 4]
    const float* fc_w  = (const float*)d_in[2];   // [1024, 12544]
    const float* fc_b  = (const float*)d_in[3];   // [1024]
    const float* gamma = (const float*)d_in[4];   // [1024]
    const float* beta  = (const float*)d_in[5];   // [1024]
    float* out = (float*)d_out;                   // [64, 1024]

    float* feat = (float*)d_ws;                   // [64, 12544] = 3.2 MB
    float* h    = feat + (size_t)B_ * KDIM_;      // [64, 1024]  = 256 KB

    // 1) RoIAlign
    {
        const int total = B_ * C_ * RES_ * RES_;  // 802816
        roi_align_kernel<<<(total + 255) / 256, 256, 0, stream>>>(x, bbox, feat);
    }
    // 2) WMMA GEMM + bias + ReLU
    {
        dim3 grid(NDIM_ / 16, B_ / 16);           // (64, 4) tiles, 1 wave each
        gemm_wmma_kernel<<<grid, 32, 0, stream>>>(feat, fc_w, fc_b, h);
    }
    // 3) BatchNorm over batch axis
    {
        bn_kernel<<<(NDIM_ + 255) / 256, 256, 0, stream>>>(h, gamma, beta, out);
    }
}